// PJCLoss_53412213293096
// MI455X (gfx1250) — compile-verified
//
#include <hip/hip_runtime.h>
#include <hip/hip_bf16.h>

typedef __attribute__((ext_vector_type(2))) float v2f;
typedef __attribute__((ext_vector_type(8))) float v8f;

// Wave reduction: each lane holds 2 f32 partials -> the wave's 64 values form
// a 4x16 f32 B-matrix. D = ones(16x4) x B yields all 16 column sums in D's
// VGPR0 (each 16-lane half holds the full set), then 4 xor-swizzles finish the
// wave total. Pure f32 (16x16x4_f32 shape) -> no precision loss.
__device__ __forceinline__ float wave_reduce_wmma(float acc0, float acc1) {
    v2f a = {1.0f, 1.0f};          // A = ones (16x4)
    v2f b = {acc0, acc1};          // B = 4x16 of wave partials
    v8f c = {};
    v8f d = __builtin_amdgcn_wmma_f32_16x16x4_f32(
        /*neg_a=*/false, a, /*neg_b=*/false, b,
        /*c_mod=*/(short)0, c, /*reuse_a=*/false, /*reuse_b=*/false);
    float v = d[0];                // lane n holds colsum[n & 15]
    // ds_swizzle group-of-32: offset = (xor_mask<<10) | and_mask(0x1f)
    v += __int_as_float(__builtin_amdgcn_ds_swizzle(__float_as_int(v), (1 << 10) | 0x1f));
    v += __int_as_float(__builtin_amdgcn_ds_swizzle(__float_as_int(v), (2 << 10) | 0x1f));
    v += __int_as_float(__builtin_amdgcn_ds_swizzle(__float_as_int(v), (4 << 10) | 0x1f));
    v += __int_as_float(__builtin_amdgcn_ds_swizzle(__float_as_int(v), (8 << 10) | 0x1f));
    return v;                      // wave total (16 distinct colsums summed)
}

// Pass 1: gather + squared-diff + per-block partial sum.
// Element e: rec[(size_t)e * D + idx[e / HW]] vs img[e].
// Requires perBlock (2048) | HW so the batch index is block-uniform.
__global__ void __launch_bounds__(256)
pjc_loss_partial(const float* __restrict__ rec,
                 const float* __restrict__ img,
                 const int*   __restrict__ idx,
                 float* __restrict__ partial,
                 int N, int HW, int D) {
    const int tid  = threadIdx.x;
    const int base = blockIdx.x * 2048;
    const int b    = base / HW;                  // scalar (blockIdx is scalar)
    const int s    = idx[b];                     // single s_load, hoisted
    const float* __restrict__ recs = rec + (size_t)s;

    // Phase 1: issue all 16 loads back-to-back (max MLP), no waits between.
    float r[8], m[8];
#pragma unroll
    for (int j = 0; j < 8; ++j) {
        int e = base + j * 256 + tid;
        e = (e < N) ? e : (N - 1);               // clamp: loads stay in bounds
        r[j] = __builtin_nontemporal_load(&recs[(size_t)e * (size_t)D]);
        m[j] = __builtin_nontemporal_load(&img[e]);
    }

    // Phase 2: accumulate (tail duplicates masked to zero; no divergence).
    float acc0 = 0.0f, acc1 = 0.0f;
#pragma unroll
    for (int j = 0; j < 8; ++j) {
        float df = r[j] - m[j];
        float sq = df * df;
        sq = ((base + j * 256 + tid) < N) ? sq : 0.0f;
        if (j & 1) acc1 += sq; else acc0 += sq;
    }

    // All 256 threads converged: EXEC all-1s for the WMMA.
    float wsum = wave_reduce_wmma(acc0, acc1);

    __shared__ float smem[8];
    const int wave = tid >> 5, lane = tid & 31;
    if (lane == 0) smem[wave] = wsum;
    __syncthreads();
    if (tid == 0) {
        float t = 0.0f;
#pragma unroll
        for (int w = 0; w < 8; ++w) t += smem[w];  // fixed order: deterministic
        partial[blockIdx.x] = t;
    }
}

// Pass 2: single wave, fixed-order accumulation of block partials, same
// WMMA+swizzle tree, scale by 1/N. Deterministic (no float atomics).
__global__ void __launch_bounds__(32)
pjc_loss_final(const float* __restrict__ partial, int nPartial,
               float* __restrict__ out, float invN) {
    const int lane = threadIdx.x;
    float acc0 = 0.0f, acc1 = 0.0f;
    for (int i = lane; i < nPartial; i += 32) {
        float p = partial[i];
        if ((i >> 5) & 1) acc1 += p; else acc0 += p;   // fixed per-lane order
    }
    float total = wave_reduce_wmma(acc0, acc1);
    if (lane == 0) out[0] = total * invN;
}

extern "C" void kernel_launch(void* const* d_in, const int* in_sizes, int n_in,
                              void* d_out, int out_size, void* d_ws, size_t ws_size,
                              hipStream_t stream) {
    const float* rec = (const float*)d_in[0];   // [B,H,W,D] f32
    const float* img = (const float*)d_in[1];   // [B,H,W]   f32
    const int*   idx = (const int*)d_in[2];     // [B]       int
    float*       out = (float*)d_out;

    const int N  = in_sizes[1];                 // B*H*W   = 1,048,576
    const int D  = (int)((long long)in_sizes[0] / (long long)in_sizes[1]); // 256
    const int B  = in_sizes[2];                 // 16
    const int HW = N / B;                       // 65,536 (2048 | HW holds)

    const int threads = 256;
    const int perBlock = threads * 8;           // 2048 elements per block
    const int blocks = (N + perBlock - 1) / perBlock;   // 512

    float* partial = (float*)d_ws;              // blocks * 4 B = 2 KB scratch

    pjc_loss_partial<<<blocks, threads, 0, stream>>>(rec, img, idx, partial, N, HW, D);
    pjc_loss_final<<<1, 32, 0, stream>>>(partial, blocks, out, 1.0f / (float)N);
}